// SAA_14697378087265
// MI455X (gfx1250) — compile-verified
//
#include <hip/hip_runtime.h>

typedef float v2f __attribute__((ext_vector_type(2)));
typedef float v8f __attribute__((ext_vector_type(8)));

#define NH 128          // heads per group (GEMM K and M dims)
#define MG 32           // groups
#define TT 4096         // time dim (GEMM N dim)
#define BB 8            // batch
#define T_CHUNK 512     // t columns per block
#define T_TILE 64       // t columns per LDS tile
#define N_JT 4          // 16-col WMMA tiles per LDS tile
#define LDS_DW (N_JT * 32 * 64)   // 8192 dwords = 32 KB, fragment-ordered

// ---------------------------------------------------------------------------
// Prep: fold depthwise affine into the pointwise weights.
//   Wf[m][o][n] = pw_w[m,o,n] * lw_w[m,n]
//   Bf[m][o]    = pw_b[m,o] + sum_n pw_w[m,o,n] * lw_b[m,n]
// ---------------------------------------------------------------------------
__global__ __launch_bounds__(128) void fold_weights(
    const float* __restrict__ lw_w, const float* __restrict__ lw_b,
    const float* __restrict__ pw_w, const float* __restrict__ pw_b,
    float* __restrict__ Wf, float* __restrict__ Bf) {
  const int row = blockIdx.x;           // m*128 + o
  const int m = row >> 7;
  const int n = threadIdx.x;            // 0..127

  const float w = pw_w[(size_t)row * NH + n];
  Wf[(size_t)row * NH + n] = w * lw_w[m * NH + n];

  __shared__ float red[NH];
  red[n] = w * lw_b[m * NH + n];
  __syncthreads();
  #pragma unroll
  for (int st = 64; st > 0; st >>= 1) {
    if (n < st) red[n] += red[n + st];
    __syncthreads();
  }
  if (n == 0) Bf[row] = red[0] + pw_b[row];
}

// ---------------------------------------------------------------------------
// Main GEMM: Out[b,m,o,t] = sum_n Wf[m,o,n] * X[b,m,n,t] + Bf[m,o]
// Block: 256 threads (8 waves); wave w owns o-tile [16w, 16w+16).
// X tiles (128x64) staged in LDS pre-swizzled into WMMA B-fragment order:
//   element (n,t) -> dword ((j*32+s)*64 + 2*lane + dw)
//   with s=n>>2, dw=n&1, lane=(t&15)+16*((n&3)>>1), j=(t%64)>>4
// so each B fragment is ONE ds_load_b64 into a contiguous pair (no movs),
// and all 128 loads per tile share a single address VGPR (lane*8) with
// immediate offsets. Compute reads are bank-conflict free.
// ---------------------------------------------------------------------------
__global__ __launch_bounds__(256) void gemm_kernel(
    const float* __restrict__ X, const float* __restrict__ Wf,
    const float* __restrict__ Bf, float* __restrict__ Out) {
  __shared__ float Xs[LDS_DW];

  const int tchunk = blockIdx.x;        // 0..7
  const int m      = blockIdx.y;        // 0..31
  const int b      = blockIdx.z;        // 0..7
  const int tid    = threadIdx.x;
  const int wave   = tid >> 5;          // 0..7
  const int lane   = tid & 31;
  const int lhalf  = lane >> 4;
  const int l16    = lane & 15;

  const int o_base = wave * 16;
  const size_t bm_base = (size_t)(b * MG + m) * NH * (size_t)TT;

  // --- A fragments (32-bit A-matrix 16x4 layout): lane0-15 M=l16,
  //     VGPR0 K=4s(+2 upper half), VGPR1 K+1. Held for the whole chunk.
  v2f a[32];
  {
    const float* wrow = Wf + (size_t)(m * NH + o_base + l16) * NH + 2 * lhalf;
    #pragma unroll
    for (int s = 0; s < 32; ++s) {
      a[s][0] = wrow[4 * s + 0];
      a[s][1] = wrow[4 * s + 1];
    }
  }

  // --- Bias per accumulator VGPR r (D row M = r + 8*lhalf).
  float bias[8];
  #pragma unroll
  for (int r = 0; r < 8; ++r)
    bias[r] = Bf[m * NH + o_base + lhalf * 8 + r];

  const int t_start = tchunk * T_CHUNK;

  // --- Software-pipelined global->reg->LDS staging.
  float4 cur[8];

  // coalesced tile load: f4 index -> n = f4>>4, t offset = (f4&15)*4
  #pragma unroll
  for (int i = 0; i < 8; ++i) {
    const int f4 = tid + i * 256;
    const int n = f4 >> 4, tt4 = (f4 & 15) * 4;
    cur[i] = *(const float4*)(X + bm_base + (size_t)n * TT + t_start + tt4);
  }

  for (int tc = 0; tc < T_CHUNK; tc += T_TILE) {
    __syncthreads();  // previous compute done reading LDS

    // scatter-store current tile into fragment order
    #pragma unroll
    for (int i = 0; i < 8; ++i) {
      const int f4 = tid + i * 256;
      const int n = f4 >> 4, tt4 = (f4 & 15) * 4;
      const int s = n >> 2, half = (n >> 1) & 1, dw = n & 1;
      const int j = tt4 >> 4, L0 = tt4 & 15;
      const int base = (j * 32 + s) * 64 + 32 * half + dw;
      Xs[base + 2 * (L0 + 0)] = cur[i].x;
      Xs[base + 2 * (L0 + 1)] = cur[i].y;
      Xs[base + 2 * (L0 + 2)] = cur[i].z;
      Xs[base + 2 * (L0 + 3)] = cur[i].w;
    }

    // issue next tile's global loads so HBM latency overlaps compute
    const bool has_next = (tc + T_TILE) < T_CHUNK;
    float4 nxt[8];
    if (has_next) {
      const int t0n = t_start + tc + T_TILE;
      #pragma unroll
      for (int i = 0; i < 8; ++i) {
        const int f4 = tid + i * 256;
        const int n = f4 >> 4, tt4 = (f4 & 15) * 4;
        nxt[i] = *(const float4*)(X + bm_base + (size_t)n * TT + t0n + tt4);
      }
    }

    __syncthreads();  // LDS tile visible to all waves

    // --- 4 independent accumulator chains, interleaved (s-outer, j-inner)
    // so same-acc WMMAs are 4 apart (covers WMMA->WMMA RAW hazard window).
    v8f acc[N_JT] = {};
    const float* bptr = Xs + lane * 2;
    #pragma unroll
    for (int s = 0; s < 32; ++s) {
      #pragma unroll
      for (int j = 0; j < N_JT; ++j) {
        const v2f bf = *(const v2f*)(bptr + (j * 32 + s) * 64);
        acc[j] = __builtin_amdgcn_wmma_f32_16x16x4_f32(
            false, a[s], false, bf, (short)0, acc[j], false, false);
      }
    }

    // --- store with bias; streamed output -> non-temporal
    #pragma unroll
    for (int j = 0; j < N_JT; ++j) {
      float* orow =
          Out + bm_base + (size_t)o_base * TT + t_start + tc + j * 16 + l16;
      #pragma unroll
      for (int r = 0; r < 8; ++r) {
        __builtin_nontemporal_store(
            acc[j][r] + bias[r], orow + (size_t)(r + 8 * lhalf) * TT);
      }
    }

    if (has_next) {
      #pragma unroll
      for (int i = 0; i < 8; ++i) cur[i] = nxt[i];
    }
  }
}

extern "C" void kernel_launch(void* const* d_in, const int* in_sizes, int n_in,
                              void* d_out, int out_size, void* d_ws, size_t ws_size,
                              hipStream_t stream) {
  const float* H_g  = (const float*)d_in[0];
  const float* lw_w = (const float*)d_in[1];
  const float* lw_b = (const float*)d_in[2];
  const float* pw_w = (const float*)d_in[3];
  const float* pw_b = (const float*)d_in[4];

  float* Wf = (float*)d_ws;                 // 32*128*128 f32 = 2 MB
  float* Bf = Wf + (size_t)MG * NH * NH;    // 32*128 f32

  fold_weights<<<MG * NH, NH, 0, stream>>>(lw_w, lw_b, pw_w, pw_b, Wf, Bf);

  dim3 grid(TT / T_CHUNK, MG, BB);          // (8, 32, 8) = 2048 blocks
  gemm_kernel<<<grid, 256, 0, stream>>>(H_g, Wf, Bf, (float*)d_out);
}